// BFS_Refine_64682207478385
// MI455X (gfx1250) — compile-verified
//
#include <hip/hip_runtime.h>

#define N_NODES 100000
#define N_EDGES 1600000
#define D       128
#define DT      64

typedef __attribute__((ext_vector_type(2))) float v2f;
typedef __attribute__((ext_vector_type(8))) float v8f;

// ---------------------------------------------------------------- build xc0
__global__ void build_xc_kernel(const float* __restrict__ x, float* __restrict__ xc) {
    int t = blockIdx.x * blockDim.x + threadIdx.x;
    if (t >= N_NODES * D) return;
    int n = t >> 7, c = t & 127;
    xc[t] = (c < 127) ? x[n * 127 + c] : 0.0f;   // color channel starts at 0
}

// ------------------------------------------- edge scatter-add of 128-wide feats
// one lane per (edge, 4-float chunk): float4 gather from xc[src], native f32 atomics to agg[dst]
__global__ void scatter_feat_kernel(const float* __restrict__ xc,
                                    const int* __restrict__ src,
                                    const int* __restrict__ dst,
                                    float* __restrict__ agg) {
    int t = blockIdx.x * blockDim.x + threadIdx.x;        // E*32 = 51.2M < 2^31
    if (t >= N_EDGES * 32) return;
    int e = t >> 5, l = t & 31;
    int s = src[e], d = dst[e];
    const float4 v = ((const float4*)xc)[s * 32 + l];
    float* o = agg + d * D + l * 4;
    unsafeAtomicAdd(o + 0, v.x); unsafeAtomicAdd(o + 1, v.y);
    unsafeAtomicAdd(o + 2, v.z); unsafeAtomicAdd(o + 3, v.w);
}

// scatter-add of the extra (mask) channel for layer 1
__global__ void scatter_ext_kernel(const float* __restrict__ ev,
                                   const int* __restrict__ src,
                                   const int* __restrict__ dst,
                                   float* __restrict__ ea) {
    int e = blockIdx.x * blockDim.x + threadIdx.x;
    if (e >= N_EDGES) return;
    unsafeAtomicAdd(&ea[dst[e]], ev[src[e]]);
}

// ------------------------------------------------------- fused GIN MLP (WMMA)
// h = sigmoid(alpha) * ( relu((xc+agg) @ W1 + b1 + extrank1) @ W2 + b2 ), in-place into xc.
// 256 threads = 8 waves; block handles 16 rows; wave w owns output cols [16w, 16w+16).
// W1/W2 (64KB each) staged into LDS once per block via global_load_async_to_lds_b128.
// fp32 WMMA 16x16x4: A[m][k] -> lane (k/2)*16+m, vgpr k%2 ; B[k][n] -> lane (k/2)*16+n, vgpr k%2 ;
// C[m][n] -> lane (m/8)*16+n, vgpr m%8.
__global__ __launch_bounds__(256) void mlp_kernel(
        float* __restrict__ xc, const float* __restrict__ agg,
        const float* __restrict__ W1, const float* __restrict__ b1,
        const float* __restrict__ W2, const float* __restrict__ b2,
        const float* __restrict__ alpha,
        const float* __restrict__ extVec, const float* __restrict__ extAgg,
        int hasExt) {
    __shared__ __align__(16) float w1s[D * D];      // 64 KB
    __shared__ __align__(16) float w2s[D * D];      // 64 KB
    __shared__ float inT[16][D + 4];
    __shared__ float actT[16][D + 4];
    __shared__ float extRow[16];

    const int tid  = threadIdx.x;
    const int tile = blockIdx.x;                 // N = 6250 * 16 exactly
    const int lane = tid & 31;
    const int wv   = tid >> 5;
    const int i    = lane & 15;
    const int hh   = lane >> 4;
    const int colg = wv * 16 + i;

    // --- async DMA of both weight matrices into LDS (16B per lane per op) ---
    {
        unsigned l1 = (unsigned)(uintptr_t)&w1s[0];
        unsigned l2 = (unsigned)(uintptr_t)&w2s[0];
        #pragma unroll
        for (int t = 0; t < 16; ++t) {
            int cI = tid + 256 * t;              // 4096 16B chunks per matrix
            asm volatile("global_load_async_to_lds_b128 %0, %1, off"
                         :: "v"(l1 + (unsigned)cI * 16u),
                            "v"((const void*)(W1 + cI * 4)) : "memory");
            asm volatile("global_load_async_to_lds_b128 %0, %1, off"
                         :: "v"(l2 + (unsigned)cI * 16u),
                            "v"((const void*)(W2 + cI * 4)) : "memory");
        }
    }

    // --- input tile (xc + agg) into LDS ---
    #pragma unroll
    for (int rep = 0; rep < 8; ++rep) {          // 2048 elems / 256 threads
        int lin = rep * 256 + tid;
        int r = lin >> 7, c = lin & 127;
        int row = tile * 16 + r;
        inT[r][c] = xc[row * D + c] + agg[row * D + c];
    }
    if (tid < 16) {
        int row = tile * 16 + tid;
        extRow[tid] = hasExt ? (extVec[row] + extAgg[row]) : 0.0f;
    }
    asm volatile("s_wait_asynccnt 0x0" ::: "memory");
    __syncthreads();

    // GEMM1: pre = (xc+agg) @ W1
    v8f acc = {};
    #pragma unroll 4
    for (int kk = 0; kk < D; kk += 4) {
        int ka = kk + 2 * hh;
        v2f a; a.x = inT[i][ka];             a.y = inT[i][ka + 1];
        v2f b; b.x = w1s[ka * D + colg];     b.y = w1s[(ka + 1) * D + colg];
        acc = __builtin_amdgcn_wmma_f32_16x16x4_f32(false, a, false, b,
                                                    (short)0, acc, false, false);
    }
    {   // bias + extra-channel rank-1 term (layer 1: channel 128 of W1) + ReLU
        float bb = b1[colg];
        float we = hasExt ? W1[D * D + colg] : 0.0f;
        #pragma unroll
        for (int v = 0; v < 8; ++v) {
            int m = v + 8 * hh;
            float p = acc[v] + bb + extRow[m] * we;
            actT[m][colg] = p > 0.0f ? p : 0.0f;
        }
    }
    __syncthreads();

    // GEMM2: y = act @ W2
    v8f acc2 = {};
    #pragma unroll 4
    for (int kk = 0; kk < D; kk += 4) {
        int ka = kk + 2 * hh;
        v2f a; a.x = actT[i][ka];            a.y = actT[i][ka + 1];
        v2f b; b.x = w2s[ka * D + colg];     b.y = w2s[(ka + 1) * D + colg];
        acc2 = __builtin_amdgcn_wmma_f32_16x16x4_f32(false, a, false, b,
                                                     (short)0, acc2, false, false);
    }
    {
        float g  = 1.0f / (1.0f + __expf(-alpha[0]));
        float bb = b2[colg];
        #pragma unroll
        for (int v = 0; v < 8; ++v) {
            int m = v + 8 * hh;
            xc[(tile * 16 + m) * D + colg] = g * (acc2[v] + bb);
        }
    }
}

// ---------------------------------------------------------------- color logic
__global__ void bincount_kernel(const int* __restrict__ col,
                                int* __restrict__ counts, int* __restrict__ maxcol) {
    __shared__ int h[DT];
    __shared__ int mx;
    if (threadIdx.x < DT) h[threadIdx.x] = 0;
    if (threadIdx.x == 0) mx = 0;
    __syncthreads();
    int t = blockIdx.x * blockDim.x + threadIdx.x;
    if (t < N_NODES) {
        int c = col[t];
        if (c >= 0 && c < DT) atomicAdd(&h[c], 1);
        atomicMax(&mx, c);
    }
    __syncthreads();
    if (threadIdx.x < DT && h[threadIdx.x]) atomicAdd(&counts[threadIdx.x], h[threadIdx.x]);
    if (threadIdx.x == 0) atomicMax(maxcol, mx);
}

__global__ void splitter_kernel(const int* __restrict__ counts, const int* __restrict__ maxcol,
                                int* __restrict__ cid, int* __restrict__ newc) {
    __shared__ int m;
    if (threadIdx.x == 0) m = DT;
    __syncthreads();
    if (threadIdx.x < DT && counts[threadIdx.x] > 1) atomicMin(&m, (int)threadIdx.x);
    __syncthreads();
    if (threadIdx.x == 0) { *cid = (m == DT) ? 0 : m; *newc = *maxcol + 1; }
}

__global__ void mask_recolor_kernel(int* __restrict__ col,
                                    const int* __restrict__ cid, const int* __restrict__ newc,
                                    float* __restrict__ extVec, int* __restrict__ maskSum) {
    __shared__ int cnt;
    if (threadIdx.x == 0) cnt = 0;
    __syncthreads();
    int t = blockIdx.x * blockDim.x + threadIdx.x;
    int mloc = 0;
    if (t < N_NODES) {
        int m = (col[t] == *cid);
        extVec[t] = m ? 1.0f : 0.0f;
        if (m) { col[t] = *newc; mloc = 1; }
    }
    atomicAdd(&cnt, mloc);
    __syncthreads();
    if (threadIdx.x == 0 && cnt) atomicAdd(maskSum, cnt);
}

// masked neighbor-color histogram over edges (uses post-recolor col, pre-recolor mask)
__global__ void sig_kernel(const int* __restrict__ src, const int* __restrict__ dst,
                           const int* __restrict__ col, const float* __restrict__ mask,
                           int* __restrict__ sig) {
    __shared__ int h[DT];
    if (threadIdx.x < DT) h[threadIdx.x] = 0;
    __syncthreads();
    int t = blockIdx.x * blockDim.x + threadIdx.x;
    if (t < N_EDGES && mask[dst[t]] != 0.0f) {
        int c = col[src[t]];
        if (c >= 0 && c < DT) atomicAdd(&h[c], 1);
    }
    __syncthreads();
    if (threadIdx.x < DT && h[threadIdx.x]) atomicAdd(&sig[threadIdx.x], h[threadIdx.x]);
}

__global__ void trace_update_kernel(float* __restrict__ trace, const int* __restrict__ sig,
                                    const int* __restrict__ maskSum) {
    int c = threadIdx.x;
    if (c < DT) {
        float sm = (float)sig[c] / (float)(*maskSum);
        trace[c] = (trace[c] + sm) * 0.5f;
    }
}

__global__ void finalize_kernel(const float* __restrict__ trace,
                                const float* __restrict__ a0, const float* __restrict__ a1,
                                float* __restrict__ out) {
    int t = threadIdx.x;
    if (t < DT)           out[t] = trace[t];
    else if (t == DT)     out[DT]     = 1.0f / (1.0f + __expf(-a0[0]));
    else if (t == DT + 1) out[DT + 1] = 1.0f / (1.0f + __expf(-a1[0]));
}

// ================================================================= launcher
extern "C" void kernel_launch(void* const* d_in, const int* in_sizes, int n_in,
                              void* d_out, int out_size, void* d_ws, size_t ws_size,
                              hipStream_t stream) {
    const float* x   = (const float*)d_in[0];
    const int*   ei  = (const int*)d_in[1];
    const int*   src = ei;
    const int*   dst = ei + N_EDGES;
    const float* W1[2]    = { (const float*)d_in[2], (const float*)d_in[7] };
    const float* b1[2]    = { (const float*)d_in[3], (const float*)d_in[8] };
    const float* W2[2]    = { (const float*)d_in[4], (const float*)d_in[9] };
    const float* b2[2]    = { (const float*)d_in[5], (const float*)d_in[10] };
    const float* alpha[2] = { (const float*)d_in[6], (const float*)d_in[11] };
    float* out = (float*)d_out;

    // workspace partition (256B aligned)
    char* p = (char*)d_ws;
    size_t featB = (size_t)N_NODES * D * sizeof(float);
    float* xc     = (float*)p;                 p += featB;
    float* agg    = (float*)p;                 p += featB;
    float* extVec = (float*)p;                 p += ((size_t)N_NODES * 4 + 255) & ~255ull;
    float* extAgg = (float*)p;                 p += ((size_t)N_NODES * 4 + 255) & ~255ull;
    int*   col    = (int*)p;                   p += ((size_t)N_NODES * 4 + 255) & ~255ull;
    float* trace  = (float*)p;                 p += 256;
    int*   scal   = (int*)p;                   // counts[64] | sig[64] | maskSum | maxcol | cid | newc
    int* counts  = scal;
    int* sig     = scal + 64;
    int* maskSum = scal + 128;
    int* maxcol  = scal + 129;
    int* cid     = scal + 130;
    int* newc    = scal + 131;

    hipMemsetAsync(col, 0, (size_t)N_NODES * 4, stream);
    hipMemsetAsync(trace, 0, DT * sizeof(float), stream);

    build_xc_kernel<<<(N_NODES * D + 255) / 256, 256, 0, stream>>>(x, xc);

    for (int layer = 0; layer < 2; ++layer) {
        hipMemsetAsync(agg, 0, featB, stream);
        hipMemsetAsync(scal, 0, 132 * sizeof(int), stream);

        scatter_feat_kernel<<<(N_EDGES * 32 + 255) / 256, 256, 0, stream>>>(xc, src, dst, agg);
        if (layer == 1) {
            hipMemsetAsync(extAgg, 0, (size_t)N_NODES * 4, stream);
            scatter_ext_kernel<<<(N_EDGES + 255) / 256, 256, 0, stream>>>(extVec, src, dst, extAgg);
        }

        mlp_kernel<<<N_NODES / 16, 256, 0, stream>>>(xc, agg,
                                                     W1[layer], b1[layer], W2[layer], b2[layer],
                                                     alpha[layer], extVec, extAgg, layer == 1 ? 1 : 0);

        bincount_kernel<<<(N_NODES + 255) / 256, 256, 0, stream>>>(col, counts, maxcol);
        splitter_kernel<<<1, DT, 0, stream>>>(counts, maxcol, cid, newc);
        mask_recolor_kernel<<<(N_NODES + 255) / 256, 256, 0, stream>>>(col, cid, newc, extVec, maskSum);
        sig_kernel<<<(N_EDGES + 255) / 256, 256, 0, stream>>>(src, dst, col, extVec, sig);
        trace_update_kernel<<<1, DT, 0, stream>>>(trace, sig, maskSum);
    }

    finalize_kernel<<<1, 128, 0, stream>>>(trace, alpha[0], alpha[1], out);
}